// BiaffineLabeler_43516608643697
// MI455X (gfx1250) — compile-verified
//
#include <hip/hip_runtime.h>

// ---------------------------------------------------------------------------
// Biaffine dependency labeler for MI455X (gfx1250, wave32, WMMA + TDM).
//   dep_label  = dep  @ dep_W^T  + dep_b      [4096 x 512]
//   head_label = head @ head_W^T + head_b     [4098 x 512]
//   sel[t]     = head_label[b, head_idx[b,t]] [4096 x 512]
//   logits[t,n]= dep_label[t] . (W[n] @ sel[t]) + bias[n]
// Heavy math on V_WMMA_F32_16X16X4_F32 (exact fp32); W slabs staged into LDS
// by the Tensor Data Mover with double buffering (TENSORcnt-tracked).
// ---------------------------------------------------------------------------

typedef float        v2f  __attribute__((ext_vector_type(2)));
typedef float        v8f  __attribute__((ext_vector_type(8)));
typedef unsigned int u32x4 __attribute__((ext_vector_type(4)));
typedef int          i32x4 __attribute__((ext_vector_type(4)));
typedef int          i32x8 __attribute__((ext_vector_type(8)));

#define D_IN     1024
#define D_LAB    512
#define N_LABELS 50
#define B_DIM    2
#define T_DIM    2048
#define NTOK     (B_DIM * T_DIM)        // 4096
#define NHEAD    (B_DIM * (T_DIM + 1))  // 4098

#if defined(__HIP_DEVICE_COMPILE__) && __has_builtin(__builtin_amdgcn_tensor_load_to_lds)
#define HAVE_TDM 1
#else
#define HAVE_TDM 0
#endif

__device__ __forceinline__ v8f wmma_f32_k4(v2f a, v2f b, v8f c) {
  // D = A(16x4 f32) * B(4x16 f32) + C(16x16 f32)
  return __builtin_amdgcn_wmma_f32_16x16x4_f32(
      false, a, false, b, (short)0, c, false, false);
}

// Issue a TDM load of a 512x16 fp32 slab (row stride 512 floats) into LDS at
// byte offset lds_off, laid out row-major [k][16].  2D descriptor, data_size=4B.
__device__ __forceinline__ void tdm_issue_slab(unsigned lds_off, const float* gsrc) {
#if HAVE_TDM
  const unsigned long long ga = (unsigned long long)(uintptr_t)gsrc;
  u32x4 g0;
  g0[0] = 1u;                                   // count=1 (valid), no gather
  g0[1] = lds_off;                              // LDS byte address (Wlds @ offset 0)
  g0[2] = (unsigned)ga;                         // global_addr[31:0]
  g0[3] = (unsigned)((ga >> 32) & 0x1FFFFFFu)   // global_addr[56:32]
        | (2u << 30);                           // type = 2 ("image")
  i32x8 g1;
  g1[0] = 0x00020000;                           // wg_mask=0, data_size=2 (4B)
  g1[1] = (int)(16u << 16);                     // tensor_dim0 = 16  ([79:48] lo)
  g1[2] = (int)(512u << 16);                    // tensor_dim1 = 512 ([111:80] lo)
  g1[3] = (int)(16u << 16);                     // tile_dim0 = 16    ([127:112])
  g1[4] = 512;                                  // tile_dim1 = 512   ([143:128])
  g1[5] = 512;                                  // tensor_dim0_stride = 512
  g1[6] = 0;
  g1[7] = 0;
  i32x4 g2 = {0, 0, 0, 0};
  i32x4 g3 = {0, 0, 0, 0};
#if defined(__clang_major__) && __clang_major__ >= 23
  i32x8 g4 = {0, 0, 0, 0, 0, 0, 0, 0};
  __builtin_amdgcn_tensor_load_to_lds(g0, g1, g2, g3, g4, 0);
#else
  __builtin_amdgcn_tensor_load_to_lds(g0, g1, g2, g3, 0);
#endif
#else
  (void)lds_off; (void)gsrc;
#endif
}

__device__ __forceinline__ void tdm_wait_le1() {
#if defined(__HIP_DEVICE_COMPILE__) && __has_builtin(__builtin_amdgcn_s_wait_tensorcnt)
  __builtin_amdgcn_s_wait_tensorcnt(1);
#endif
}
__device__ __forceinline__ void tdm_wait_le0() {
#if defined(__HIP_DEVICE_COMPILE__) && __has_builtin(__builtin_amdgcn_s_wait_tensorcnt)
  __builtin_amdgcn_s_wait_tensorcnt(0);
#endif
}

// ---------------------------------------------------------------------------
// Projection: out[r,e] = sum_d X[r,d] * Wp[e,d] + bp[e]
// Block = 128 (4 waves); wave w owns row-tile blockIdx.x*4+w and a 16x64
// output strip (4 accumulators) so each A fragment feeds 4 WMMAs.
// grid = (ceil(rtiles/4), D_LAB/64)
// ---------------------------------------------------------------------------
__global__ void __launch_bounds__(128)
proj16(const float* __restrict__ X, const float* __restrict__ Wp,
       const float* __restrict__ bp, float* __restrict__ out, int nrows)
{
  const int lane = threadIdx.x & 31;
  const int wave = threadIdx.x >> 5;
  const int m = lane & 15;
  const int g = lane >> 4;
  const int rbase = (blockIdx.x * 4 + wave) * 16;
  const int ebase = blockIdx.y * 64;

  int rowA = rbase + m;
  if (rowA > nrows - 1) rowA = nrows - 1;   // clamp partial last tile

  const float* aptr = X + (size_t)rowA * D_IN + 2 * g;
  const float* bptr0 = Wp + (size_t)(ebase +  0 + m) * D_IN + 2 * g;
  const float* bptr1 = Wp + (size_t)(ebase + 16 + m) * D_IN + 2 * g;
  const float* bptr2 = Wp + (size_t)(ebase + 32 + m) * D_IN + 2 * g;
  const float* bptr3 = Wp + (size_t)(ebase + 48 + m) * D_IN + 2 * g;

  v8f acc0 = {}, acc1 = {}, acc2 = {}, acc3 = {};
#pragma unroll 4
  for (int k = 0; k < D_IN; k += 4) {
    v2f a  = *(const v2f*)(aptr + k);
    v2f b0 = *(const v2f*)(bptr0 + k);
    v2f b1 = *(const v2f*)(bptr1 + k);
    v2f b2 = *(const v2f*)(bptr2 + k);
    v2f b3 = *(const v2f*)(bptr3 + k);
    acc0 = wmma_f32_k4(a, b0, acc0);
    acc1 = wmma_f32_k4(a, b1, acc1);
    acc2 = wmma_f32_k4(a, b2, acc2);
    acc3 = wmma_f32_k4(a, b3, acc3);
  }

  const float bv0 = bp[ebase +  0 + m];
  const float bv1 = bp[ebase + 16 + m];
  const float bv2 = bp[ebase + 32 + m];
  const float bv3 = bp[ebase + 48 + m];

  if (rbase + 16 <= nrows) {          // fast path: full tile
#pragma unroll
    for (int j = 0; j < 8; ++j) {
      float* o = out + (size_t)(rbase + j + 8 * g) * D_LAB + ebase + m;
      o[ 0] = acc0[j] + bv0;
      o[16] = acc1[j] + bv1;
      o[32] = acc2[j] + bv2;
      o[48] = acc3[j] + bv3;
    }
  } else {
#pragma unroll
    for (int j = 0; j < 8; ++j) {
      const int row = rbase + j + 8 * g;
      if (row < nrows) {
        float* o = out + (size_t)row * D_LAB + ebase + m;
        o[ 0] = acc0[j] + bv0;
        o[16] = acc1[j] + bv1;
        o[32] = acc2[j] + bv2;
        o[48] = acc3[j] + bv3;
      }
    }
  }
}

// ---------------------------------------------------------------------------
// Gather: sel[token,:] = head_label[b*(T+1) + head_idx[token], :]
// ---------------------------------------------------------------------------
__global__ void __launch_bounds__(128)
gather_sel(const float* __restrict__ head_label, const int* __restrict__ hidx,
           float* __restrict__ sel)
{
  const int token = blockIdx.x;
  const int b = token / T_DIM;
  const int idx = hidx[token];
  const float4* src = (const float4*)(head_label +
                      (size_t)(b * (T_DIM + 1) + idx) * D_LAB);
  float4* dst = (float4*)(sel + (size_t)token * D_LAB);
  dst[threadIdx.x] = src[threadIdx.x];
}

// ---------------------------------------------------------------------------
// Biaffine: logits[t,n] = sum_e ( sum_d dep_label[t,d] * W[n,d,e] ) * sel[t,e]
// grid = (NTOK/256, N_LABELS), block = 256 (8 waves).
// Each wave owns TWO 16-token tiles (one B fragment feeds 2 WMMAs).
// W[n][:, e..e+15] slabs are TDM-loaded into a 2x32KB LDS double buffer,
// laid out row-major [k][16]; issue slab et+1, s_wait_tensorcnt(1) => slab et
// has landed, so the DMA of the next slab overlaps this slab's 256 WMMAs.
// ---------------------------------------------------------------------------
__global__ void __launch_bounds__(256)
biaffine16(const float* __restrict__ dep_label, const float* __restrict__ sel,
           const float* __restrict__ Wall, const float* __restrict__ bias,
           float* __restrict__ out)
{
  // Only __shared__ object => LDS byte offset 0 (TDM descriptors rely on this).
  __shared__ float Wlds[2][D_LAB * 16];   // 64 KB double buffer

  const int tid  = threadIdx.x;
  const int lane = tid & 31;
  const int wave = tid >> 5;                      // 0..7
  const int m = lane & 15;
  const int g = lane >> 4;
  const int n = blockIdx.y;                       // label
  const int tt0 = (blockIdx.x * 8 + wave) * 2;    // first of 2 token tiles
  const int t0 = tt0 * 16;
  const int t1 = t0 + 16;

  const float* Wn  = Wall + (size_t)n * D_LAB * D_LAB;    // W[n][d][e] row-major
  const float* a0p = dep_label + (size_t)(t0 + m) * D_LAB + 2 * g;
  const float* a1p = dep_label + (size_t)(t1 + m) * D_LAB + 2 * g;

  float p0[8], p1[8];
#pragma unroll
  for (int j = 0; j < 8; ++j) { p0[j] = 0.0f; p1[j] = 0.0f; }

  for (int et = 0; et < D_LAB / 16; ++et) {
    const int cur = et & 1;

    __syncthreads();   // buf[cur] consumers from iteration et-2 are done

#if HAVE_TDM
    if (wave == 0) {
      if (et == 0)
        tdm_issue_slab(0u, Wn);                              // slab 0 -> buf 0
      if (et + 1 < D_LAB / 16) {
        tdm_issue_slab(((et + 1) & 1) ? 32768u : 0u,          // slab et+1
                       Wn + (et + 1) * 16);
        tdm_wait_le1();                                       // slab et landed
      } else {
        tdm_wait_le0();
      }
    }
#else
    {
      // Fallback: cooperative synchronous copy of slab et, [k][16] layout.
      const int e4 = (tid & 3) * 4;
      const float4* wsrc = (const float4*)(Wn + et * 16);
      for (int k = tid >> 2; k < D_LAB; k += 64) {
        float4 v = wsrc[(size_t)k * (D_LAB / 4) + (e4 >> 2)];
        *(float4*)&Wlds[cur][k * 16 + e4] = v;
      }
      if (et + 1 < D_LAB / 16)
        __builtin_prefetch(Wn + (size_t)(lane * 16) * D_LAB + (et + 1) * 16, 0, 1);
    }
#endif
    __syncthreads();   // buf[cur] holds slab et, visible to all waves

    // C(16 tok x 16 e) per tile; B fragment: lane(m,g) needs
    // W[k+2g][e=m] and W[k+2g+1][e=m] = Wlds[cur][(k+2g)*16+m], [..+16]
    v8f c0 = {}, c1 = {};
    const float* bl = &Wlds[cur][m];
#pragma unroll 8
    for (int k = 0; k < D_LAB; k += 4) {
      v2f a0 = *(const v2f*)(a0p + k);
      v2f a1 = *(const v2f*)(a1p + k);
      const int kk = (k + 2 * g) * 16;
      v2f b = { bl[kk], bl[kk + 16] };
      c0 = wmma_f32_k4(a0, b, c0);
      c1 = wmma_f32_k4(a1, b, c1);
    }

    // Fuse: partial[t] += C[t, ebase+m] * sel[t, ebase+m]
    const int ebase = et * 16;
#pragma unroll
    for (int j = 0; j < 8; ++j) {
      const float s0 = sel[(size_t)(t0 + j + 8 * g) * D_LAB + ebase + m];
      const float s1 = sel[(size_t)(t1 + j + 8 * g) * D_LAB + ebase + m];
      p0[j] += c0[j] * s0;
      p1[j] += c1[j] * s1;
    }
  }

  // Reduce over the 16 e-columns held by lanes m=0..15 of each half-wave.
#pragma unroll
  for (int j = 0; j < 8; ++j) {
    float v0 = p0[j], v1 = p1[j];
    v0 += __shfl_xor(v0, 1, 16);  v1 += __shfl_xor(v1, 1, 16);
    v0 += __shfl_xor(v0, 2, 16);  v1 += __shfl_xor(v1, 2, 16);
    v0 += __shfl_xor(v0, 4, 16);  v1 += __shfl_xor(v1, 4, 16);
    v0 += __shfl_xor(v0, 8, 16);  v1 += __shfl_xor(v1, 8, 16);
    p0[j] = v0; p1[j] = v1;
  }

  if (m == 0) {   // lane 0 -> tokens +0..7, lane 16 -> tokens +8..15
    const float bn = bias[n];
#pragma unroll
    for (int j = 0; j < 8; ++j) {
      out[(size_t)(t0 + j + 8 * g) * N_LABELS + n] = p0[j] + bn;
      out[(size_t)(t1 + j + 8 * g) * N_LABELS + n] = p1[j] + bn;
    }
  }
}

// ---------------------------------------------------------------------------
extern "C" void kernel_launch(void* const* d_in, const int* in_sizes, int n_in,
                              void* d_out, int out_size, void* d_ws, size_t ws_size,
                              hipStream_t stream)
{
  (void)in_sizes; (void)n_in; (void)out_size; (void)ws_size;

  const float* dep    = (const float*)d_in[0];   // [B,T,D]
  const float* head   = (const float*)d_in[1];   // [B,T+1,D]
  const int*   hidx   = (const int*)  d_in[2];   // [B,T]
  /* d_in[3] = mask: unused by the reference */
  const float* dep_W  = (const float*)d_in[4];   // [512,1024]
  const float* dep_b  = (const float*)d_in[5];   // [512]
  const float* head_W = (const float*)d_in[6];   // [512,1024]
  const float* head_b = (const float*)d_in[7];   // [512]
  const float* W      = (const float*)d_in[8];   // [50,512,512]
  const float* bias   = (const float*)d_in[9];   // [50]
  float* out = (float*)d_out;                    // [B,T,50]

  float* ws         = (float*)d_ws;
  float* dep_label  = ws;                              // 4096*512 f32
  float* sel        = ws + (size_t)NTOK * D_LAB;       // 4096*512 f32
  float* head_label = ws + (size_t)2 * NTOK * D_LAB;   // 4098*512 f32

  const int dep_rt  = NTOK / 16;                 // 256
  const int head_rt = (NHEAD + 15) / 16;         // 257

  proj16<<<dim3((dep_rt  + 3) / 4, D_LAB / 64), 128, 0, stream>>>(
      dep,  dep_W,  dep_b,  dep_label,  NTOK);
  proj16<<<dim3((head_rt + 3) / 4, D_LAB / 64), 128, 0, stream>>>(
      head, head_W, head_b, head_label, NHEAD);
  gather_sel<<<dim3(NTOK), 128, 0, stream>>>(head_label, hidx, sel);
  biaffine16<<<dim3(NTOK / 256, N_LABELS), 256, 0, stream>>>(
      dep_label, sel, W, bias, out);
}